// GAT_91182155694568
// MI455X (gfx1250) — compile-verified
//
#include <hip/hip_runtime.h>
#include <cstddef>

// ---------------------------------------------------------------------------
// GAT forward for MI455X (gfx1250, wave32).
//   layer sizes: N=50000, E=800000 (+N self loops), IN=128, HID=64, HEADS=4,
//   OUT=10, G=512.  GEMMs use V_WMMA_F32_16X16X4_F32 (fp32 in, fp32 acc).
//   Edge softmax: 3-pass segment softmax with float atomics (L2-resident).
// ---------------------------------------------------------------------------

#define NEG_SLOPE 0.2f

typedef float v2f __attribute__((ext_vector_type(2)));
typedef float v8f __attribute__((ext_vector_type(8)));

// ---------------- WMMA GEMM: C[M,N] = A[M,K] * B[K,N], fp32 -----------------
// Block: 128 threads = 4 waves. Each block: one 16-row M tile, 4 N tiles of 16.
// A panel (16 x K) staged in LDS (K <= 256 -> 16KB).
// Fragment layouts per CDNA5 ISA 7.12.2:
//   A 16x4:  lanes 0-15 -> {K=0,K=1}, lanes 16-31 -> {K=2,K=3}, row = lane%16
//   B 4x16:  lanes 0-15 -> rows {0,1}, lanes 16-31 -> rows {2,3}, col = lane%16
//   C 16x16: VGPR r -> row r (lanes 0-15) / row r+8 (lanes 16-31), col = lane%16
__global__ void gat_gemm_wmma(const float* __restrict__ A,
                              const float* __restrict__ B,
                              float* __restrict__ C,
                              int M, int N, int K)
{
    __shared__ float As[16 * 256];
    const int m0   = blockIdx.x * 16;
    const int wave = threadIdx.x >> 5;
    const int lane = threadIdx.x & 31;
    const int half = lane >> 4;      // 0: lanes 0-15, 1: lanes 16-31
    const int l15  = lane & 15;
    const int n0   = (blockIdx.y * (blockDim.x >> 5) + wave) * 16;

    for (int i = threadIdx.x; i < 16 * K; i += blockDim.x) {
        const int r = i / K, c = i % K;
        const int gm = m0 + r;
        As[i] = (gm < M) ? A[(size_t)gm * K + c] : 0.0f;
    }
    __syncthreads();

    v8f acc = {};
    for (int k = 0; k < K; k += 4) {
        v2f a, b;
        const int ka = k + 2 * half;
        a.x = As[l15 * K + ka + 0];
        a.y = As[l15 * K + ka + 1];
        b.x = B[(size_t)(ka + 0) * N + n0 + l15];
        b.y = B[(size_t)(ka + 1) * N + n0 + l15];
        acc = __builtin_amdgcn_wmma_f32_16x16x4_f32(
            false, a, false, b, (short)0, acc, false, false);
    }

    #pragma unroll
    for (int r = 0; r < 8; ++r) {
        const int gm = m0 + r + 8 * half;
        if (gm < M) C[(size_t)gm * N + n0 + l15] = acc[r];
    }
}

// ------------- per-(node,head) attention coefficients a_s, a_d --------------
__global__ void gat_attn_coef(const float* __restrict__ H,
                              const float* __restrict__ att_s,
                              const float* __restrict__ att_d,
                              float* __restrict__ as_, float* __restrict__ ad_,
                              int Nn, int heads, int C)
{
    const int idx = blockIdx.x * blockDim.x + threadIdx.x;
    if (idx >= Nn * heads) return;
    const int n = idx / heads, h = idx % heads;
    const float* hp = H + (size_t)n * heads * C + h * C;
    const float* sp = att_s + h * C;
    const float* dp = att_d + h * C;
    float ss = 0.f, dd = 0.f;
    #pragma unroll 8
    for (int c = 0; c < C; ++c) { const float v = hp[c]; ss += v * sp[c]; dd += v * dp[c]; }
    as_[idx] = ss; ad_[idx] = dd;
}

// ---------------- float atomic max via int/uint ordering trick --------------
__device__ __forceinline__ void atomicMaxF32(float* addr, float val)
{
    if (val >= 0.0f) atomicMax((int*)addr, __float_as_int(val));
    else             atomicMin((unsigned int*)addr, __float_as_uint(val));
}

__device__ __forceinline__ void edge_endpoints(const int* ei, int E, int e,
                                               int& s, int& d)
{
    if (e < E) { s = ei[e]; d = ei[E + e]; }
    else       { s = d = e - E; }   // self loop
}

// -------------------- pass 1: segment max of edge scores --------------------
__global__ void gat_edge_max(const int* __restrict__ ei, int E, int Etot,
                             const float* __restrict__ as_,
                             const float* __restrict__ ad_,
                             float* __restrict__ mbuf, int heads)
{
    const int e = blockIdx.x * blockDim.x + threadIdx.x;
    if (e >= Etot) return;
    int s, d; edge_endpoints(ei, E, e, s, d);
    for (int h = 0; h < heads; ++h) {
        float ev = as_[s * heads + h] + ad_[d * heads + h];
        ev = ev > 0.f ? ev : NEG_SLOPE * ev;
        atomicMaxF32(&mbuf[d * heads + h], ev);
    }
}

// ------- pass 2: alpha = exp(e - m[dst]); scatter numerator + denom ---------
// one wave per edge; lanes split channels for coalesced atomics
__global__ void gat_edge_aggregate(const int* __restrict__ ei, int E, int Etot,
                                   const float* __restrict__ as_,
                                   const float* __restrict__ ad_,
                                   const float* __restrict__ mbuf,
                                   float* __restrict__ den,
                                   float* __restrict__ accum,
                                   const float* __restrict__ H,
                                   int heads, int C)
{
    const int e = blockIdx.x * (blockDim.x >> 5) + (threadIdx.x >> 5);
    if (e >= Etot) return;
    const int lane = threadIdx.x & 31;
    int s, d; edge_endpoints(ei, E, e, s, d);
    const int HC = heads * C;
    const float* hs = H + (size_t)s * HC;
    float* acc = accum + (size_t)d * HC;
    for (int h = 0; h < heads; ++h) {
        float ev = as_[s * heads + h] + ad_[d * heads + h];
        ev = ev > 0.f ? ev : NEG_SLOPE * ev;
        const float alpha = __expf(ev - mbuf[d * heads + h]);
        if (lane == 0) atomicAdd(&den[d * heads + h], alpha);
        for (int c = lane; c < C; c += 32)
            atomicAdd(&acc[h * C + c], alpha * hs[h * C + c]);
    }
}

// ------------ pass 3: normalize by denom, + bias, ELU (in place) ------------
__global__ void gat_node_epilogue(float* __restrict__ accum,
                                  const float* __restrict__ den,
                                  const float* __restrict__ bias,
                                  int Nn, int heads, int C)
{
    const int HC = heads * C;
    const int idx = blockIdx.x * blockDim.x + threadIdx.x;
    if (idx >= Nn * HC) return;
    const int n = idx / HC, rem = idx % HC, h = rem / C;
    float v = accum[idx] / den[n * heads + h] + bias[rem];
    accum[idx] = v > 0.f ? v : (__expf(v) - 1.0f);   // ELU(alpha=1)
}

// -------------------------- global mean pool --------------------------------
__global__ void gat_pool(const float* __restrict__ h2,
                         const int* __restrict__ batch,
                         float* __restrict__ sums, float* __restrict__ cnts,
                         int Nn, int C)
{
    const int idx = blockIdx.x * blockDim.x + threadIdx.x;
    if (idx >= Nn * C) return;
    const int n = idx / C, c = idx % C;
    const int b = batch[n];
    atomicAdd(&sums[b * C + c], h2[idx]);
    if (c == 0) atomicAdd(&cnts[b], 1.0f);
}

// ---------------------- final linear: (sums/cnt) @ Wl + bl ------------------
__global__ void gat_final_linear(const float* __restrict__ sums,
                                 const float* __restrict__ cnts,
                                 const float* __restrict__ Wl,
                                 const float* __restrict__ bl,
                                 float* __restrict__ out, int G, int C, int O)
{
    const int idx = blockIdx.x * blockDim.x + threadIdx.x;
    if (idx >= G * O) return;
    const int g = idx / O, o = idx % O;
    const float inv = 1.0f / fmaxf(cnts[g], 1.0f);
    float a = bl[o];
    for (int c = 0; c < C; ++c) a += sums[g * C + c] * inv * Wl[c * O + o];
    out[idx] = a;
}

__global__ void gat_fill(float* __restrict__ p, float v, size_t n)
{
    const size_t i = (size_t)blockIdx.x * blockDim.x + threadIdx.x;
    if (i < n) p[i] = v;
}

// ---------------------------------------------------------------------------
extern "C" void kernel_launch(void* const* d_in, const int* in_sizes, int n_in,
                              void* d_out, int out_size, void* d_ws, size_t ws_size,
                              hipStream_t stream)
{
    const float* x   = (const float*)d_in[0];
    const int*   ei  = (const int*)  d_in[1];
    const int*   bat = (const int*)  d_in[2];
    const float* W1  = (const float*)d_in[3];
    const float* aS1 = (const float*)d_in[4];
    const float* aD1 = (const float*)d_in[5];
    const float* b1  = (const float*)d_in[6];
    const float* W2  = (const float*)d_in[7];
    const float* aS2 = (const float*)d_in[8];
    const float* aD2 = (const float*)d_in[9];
    const float* b2  = (const float*)d_in[10];
    const float* Wl  = (const float*)d_in[11];
    const float* bl  = (const float*)d_in[12];
    float* out = (float*)d_out;

    const int N    = in_sizes[2];          // 50000
    const int E    = in_sizes[1] / 2;      // 800000
    const int Etot = E + N;                // + self loops
    const int IN   = in_sizes[0] / N;      // 128
    const int HID  = 64, HEADS = 4, OUT = 10, G = 512;
    const int HC1  = HEADS * HID;          // 256

    // ---- workspace layout (floats) ----
    float* ws = (float*)d_ws;
    size_t off = 0;
    float* H1  = ws + off; off += (size_t)N * HC1;
    float* A1  = ws + off; off += (size_t)N * HC1;   // accum1 -> h1 activated
    float* as1 = ws + off; off += (size_t)N * HEADS;
    float* ad1 = ws + off; off += (size_t)N * HEADS;
    float* m1  = ws + off; off += (size_t)N * HEADS;
    float* dn1 = ws + off; off += (size_t)N * HEADS;
    float* H2  = ws + off; off += (size_t)N * HID;
    float* A2  = ws + off; off += (size_t)N * HID;   // accum2 -> h2 activated
    float* as2 = ws + off; off += (size_t)N;
    float* ad2 = ws + off; off += (size_t)N;
    float* m2  = ws + off; off += (size_t)N;
    float* dn2 = ws + off; off += (size_t)N;
    float* sums = ws + off; off += (size_t)G * HID;
    float* cnts = ws + off; off += (size_t)G;
    (void)ws_size; (void)n_in;

    const float NEG_INF = __builtin_inff() * -1.0f;
    const int TB = 256;
    auto blks = [](size_t n, int tb) { return (unsigned)((n + tb - 1) / tb); };

    // ===================== layer 1 =====================
    // H1 = X @ W1   [N,128]x[128,256]
    {
        dim3 grid((N + 15) / 16, HC1 / 64);   // 4 waves x 16 cols = 64 cols
        gat_gemm_wmma<<<grid, 128, 0, stream>>>(x, W1, H1, N, HC1, IN);
    }
    gat_attn_coef<<<blks((size_t)N * HEADS, TB), TB, 0, stream>>>(
        H1, aS1, aD1, as1, ad1, N, HEADS, HID);
    gat_fill<<<blks((size_t)N * HEADS, TB), TB, 0, stream>>>(m1, NEG_INF, (size_t)N * HEADS);
    gat_fill<<<blks((size_t)N * HEADS, TB), TB, 0, stream>>>(dn1, 0.f, (size_t)N * HEADS);
    gat_fill<<<blks((size_t)N * HC1, TB), TB, 0, stream>>>(A1, 0.f, (size_t)N * HC1);
    gat_edge_max<<<blks(Etot, TB), TB, 0, stream>>>(ei, E, Etot, as1, ad1, m1, HEADS);
    gat_edge_aggregate<<<blks(Etot, 8) , TB, 0, stream>>>(   // 8 waves/block, wave/edge
        ei, E, Etot, as1, ad1, m1, dn1, A1, H1, HEADS, HID);
    gat_node_epilogue<<<blks((size_t)N * HC1, TB), TB, 0, stream>>>(
        A1, dn1, b1, N, HEADS, HID);

    // ===================== layer 2 =====================
    // H2 = H1' @ W2   [N,256]x[256,64]
    {
        dim3 grid((N + 15) / 16, HID / 64);   // grid.y = 1
        gat_gemm_wmma<<<grid, 128, 0, stream>>>(A1, W2, H2, N, HID, HC1);
    }
    gat_attn_coef<<<blks(N, TB), TB, 0, stream>>>(H2, aS2, aD2, as2, ad2, N, 1, HID);
    gat_fill<<<blks(N, TB), TB, 0, stream>>>(m2, NEG_INF, (size_t)N);
    gat_fill<<<blks(N, TB), TB, 0, stream>>>(dn2, 0.f, (size_t)N);
    gat_fill<<<blks((size_t)N * HID, TB), TB, 0, stream>>>(A2, 0.f, (size_t)N * HID);
    gat_edge_max<<<blks(Etot, TB), TB, 0, stream>>>(ei, E, Etot, as2, ad2, m2, 1);
    gat_edge_aggregate<<<blks(Etot, 8), TB, 0, stream>>>(
        ei, E, Etot, as2, ad2, m2, dn2, A2, H2, 1, HID);
    gat_node_epilogue<<<blks((size_t)N * HID, TB), TB, 0, stream>>>(
        A2, dn2, b2, N, 1, HID);

    // ===================== pool + linear =====================
    gat_fill<<<blks((size_t)G * HID, TB), TB, 0, stream>>>(sums, 0.f, (size_t)G * HID);
    gat_fill<<<blks(G, TB), TB, 0, stream>>>(cnts, 0.f, (size_t)G);
    gat_pool<<<blks((size_t)N * HID, TB), TB, 0, stream>>>(A2, bat, sums, cnts, N, HID);
    gat_final_linear<<<blks((size_t)G * OUT, TB), TB, 0, stream>>>(
        sums, cnts, Wl, bl, out, G, HID, OUT);
}